// SpGraphAttentionLayer_19138374271052
// MI455X (gfx1250) — compile-verified
//
#include <hip/hip_runtime.h>
#include <math.h>

typedef __attribute__((ext_vector_type(2))) float v2f;
typedef __attribute__((ext_vector_type(8))) float v8f;

#define D_IN  128
#define D_OUT 64

// ---------------------------------------------------------------------------
// Kernel 1: CSR row pointers from sorted dst (lower_bound per node).
// ---------------------------------------------------------------------------
__global__ __launch_bounds__(256) void rowptr_kernel(const int* __restrict__ dst,
                                                     int* __restrict__ rowptr,
                                                     int n, int e) {
    int i = blockIdx.x * blockDim.x + threadIdx.x;
    if (i > n) return;
    int lo = 0, hi = e;
    while (lo < hi) {
        int mid = (lo + hi) >> 1;
        if (dst[mid] < i) lo = mid + 1; else hi = mid;
    }
    rowptr[i] = lo;  // rowptr[n] == e
}

// ---------------------------------------------------------------------------
// Kernel 2: fused projections  Hk = Xk*Wk + bk,  Hv = Xv*Wv + bv  via WMMA.
// One wave per 16-row node tile; 4 column tiles of 16 cover D_OUT=64.
// f32 WMMA: A is 16x4 (2 floats/lane), B is 4x16 (2 floats/lane), C/D 16x16.
// ---------------------------------------------------------------------------
__global__ __launch_bounds__(128) void proj_kernel(
    const float* __restrict__ Xk, const float* __restrict__ Xv,
    const float* __restrict__ Wk, const float* __restrict__ bk,
    const float* __restrict__ Wv, const float* __restrict__ bv,
    float* __restrict__ Hk, float* __restrict__ Hv, int N)
{
    const int lane  = threadIdx.x & 31;
    const int wid   = threadIdx.x >> 5;
    const int tile  = blockIdx.x * 4 + wid;
    const int tiles = (N + 15) >> 4;
    if (tile >= tiles) return;               // wave-uniform: EXEC all-1 inside
    const int b    = blockIdx.y;
    const int half = lane >> 4;              // 0: K pair {0,1}, 1: K pair {2,3}
    const int l15  = lane & 15;

    int rowA  = tile * 16 + l15;
    int rowAc = rowA < N ? rowA : N - 1;     // clamp (loads only)
    const float* xk_row = Xk + ((size_t)b * N + rowAc) * D_IN;
    const float* xv_row = Xv + ((size_t)b * N + rowAc) * D_IN;

    v8f ak[4] = {{}, {}, {}, {}};
    v8f av[4] = {{}, {}, {}, {}};

#pragma unroll 4
    for (int k = 0; k < D_IN; k += 4) {
        const int k0 = k + 2 * half;
        v2f a_k, a_v;
        a_k[0] = xk_row[k0]; a_k[1] = xk_row[k0 + 1];
        a_v[0] = xv_row[k0]; a_v[1] = xv_row[k0 + 1];
#pragma unroll
        for (int n = 0; n < 4; ++n) {
            const int col = n * 16 + l15;
            v2f b_k, b_v;
            b_k[0] = Wk[(size_t)k0 * D_OUT + col];
            b_k[1] = Wk[(size_t)(k0 + 1) * D_OUT + col];
            b_v[0] = Wv[(size_t)k0 * D_OUT + col];
            b_v[1] = Wv[(size_t)(k0 + 1) * D_OUT + col];
            ak[n] = __builtin_amdgcn_wmma_f32_16x16x4_f32(
                false, a_k, false, b_k, (short)0, ak[n], false, false);
            av[n] = __builtin_amdgcn_wmma_f32_16x16x4_f32(
                false, a_v, false, b_v, (short)0, av[n], false, false);
        }
    }

    // C/D layout: VGPR j -> row (j + 8*half), col = lane&15 within col-tile.
#pragma unroll
    for (int n = 0; n < 4; ++n) {
        const int col = n * 16 + l15;
        const float biask = bk[col];
        const float biasv = bv[col];
#pragma unroll
        for (int j = 0; j < 8; ++j) {
            const int row = tile * 16 + j + 8 * half;
            if (row < N) {
                const size_t o = ((size_t)b * N + row) * D_OUT + col;
                Hk[o] = ak[n][j] + biask;
                Hv[o] = av[n][j] + biasv;
            }
        }
    }
}

// ---------------------------------------------------------------------------
// Kernel 3: one wave32 per destination node; online softmax over its edges.
// Lane d handles feature components d and d+32.
// ---------------------------------------------------------------------------
__global__ __launch_bounds__(128) void attn_kernel(
    const float* __restrict__ hkey, const float* __restrict__ hval,
    const int* __restrict__ src, const int* __restrict__ rowptr,
    float* __restrict__ out, int N)
{
    const int lane = threadIdx.x & 31;
    const int wid  = threadIdx.x >> 5;
    const int node = blockIdx.x * 4 + wid;
    const int b    = blockIdx.y;
    if (node >= N) return;

    const float* hk = hkey + (size_t)b * N * D_OUT;
    const float* hv = hval + (size_t)b * N * D_OUT;

    const float hkd0 = hk[(size_t)node * D_OUT + lane];
    const float hkd1 = hk[(size_t)node * D_OUT + lane + 32];

    const int e0 = rowptr[node];
    const int e1 = rowptr[node + 1];

    float m = -INFINITY, s = 0.f, num0 = 0.f, num1 = 0.f;

    for (int ei = e0; ei < e1; ++ei) {
        const int sn = src[ei];
        const size_t so = (size_t)sn * D_OUT;
        // 64-wide dot product, 2 components per lane + wave reduction
        float p = hkd0 * hk[so + lane] + hkd1 * hk[so + lane + 32];
#pragma unroll
        for (int off = 16; off > 0; off >>= 1)
            p += __shfl_xor(p, off, 32);
        const float e  = p;                 // same value in all lanes
        const float mn = fmaxf(m, e);
        const float corr = expf(m - mn);    // 0 on first edge (m = -inf)
        const float w    = expf(e - mn);
        s    = s * corr + w;
        num0 = num0 * corr + w * hv[so + lane];
        num1 = num1 * corr + w * hv[so + lane + 32];
        m = mn;
    }

    const float inv = (s == 0.f) ? 0.f : 1.f / s;   // no-edge nodes -> 0
    float o0 = num0 * inv;
    float o1 = num1 * inv;
    o0 = (o0 >= 0.f) ? o0 : 0.2f * o0;              // LeakyReLU(0.2)
    o1 = (o1 >= 0.f) ? o1 : 0.2f * o1;

    float* op = out + ((size_t)b * N + node) * D_OUT;
    op[lane]      = o0;
    op[lane + 32] = o1;
}

// ---------------------------------------------------------------------------
extern "C" void kernel_launch(void* const* d_in, const int* in_sizes, int n_in,
                              void* d_out, int out_size, void* d_ws, size_t ws_size,
                              hipStream_t stream) {
    const float* X_key   = (const float*)d_in[0];
    const float* X_value = (const float*)d_in[1];
    const float* Wk      = (const float*)d_in[2];
    const float* bk      = (const float*)d_in[3];
    const float* Wv      = (const float*)d_in[4];
    const float* bv      = (const float*)d_in[5];
    const int*   src     = (const int*)d_in[6];
    const int*   dst     = (const int*)d_in[7];
    float*       out     = (float*)d_out;

    const int B  = 2;
    const int BN = in_sizes[0] / D_IN;   // B*N
    const int N  = BN / B;
    const int E  = in_sizes[6];

    // workspace layout
    float* hkey  = (float*)d_ws;
    float* hval  = hkey + (size_t)BN * D_OUT;
    int*   rowpt = (int*)(hval + (size_t)BN * D_OUT);

    {   // CSR row pointers (dst identical across batches)
        int threads = 256;
        int blocks  = (N + 1 + threads - 1) / threads;
        rowptr_kernel<<<blocks, threads, 0, stream>>>(dst, rowpt, N, E);
    }
    {   // fused WMMA projections
        const int tiles = (N + 15) >> 4;
        dim3 grid((tiles + 3) / 4, B, 1);
        proj_kernel<<<grid, dim3(128, 1, 1), 0, stream>>>(
            X_key, X_value, Wk, bk, Wv, bv, hkey, hval, N);
    }
    {   // online-softmax edge aggregation, one wave per node
        dim3 grid((N + 3) / 4, B, 1);
        attn_kernel<<<grid, dim3(128, 1, 1), 0, stream>>>(
            hkey, hval, src, rowpt, out, N);
    }
}